// HETE_GNN_53644141527347
// MI455X (gfx1250) — compile-verified
//
#include <hip/hip_runtime.h>

#define IN_F 256
#define HF   10
#define HP   16   // padded row stride for internal node-feature buffers

typedef __attribute__((ext_vector_type(2))) float v2f;
typedef __attribute__((ext_vector_type(4))) float v4f;
typedef __attribute__((ext_vector_type(8))) float v8f;

// ---------------- utility ----------------
__global__ void k_zero(float* __restrict__ p, int n) {
    int i = blockIdx.x * blockDim.x + threadIdx.x;
    if (i < n) p[i] = 0.0f;
}

// deg[dst] += 1 over real edges (self-loop handled as +1 in k_dinv)
__global__ void k_deg(const int* __restrict__ dst, float* __restrict__ deg, int n_edges) {
    int e = blockIdx.x * blockDim.x + threadIdx.x;
    if (e >= n_edges) return;
    __builtin_prefetch(&dst[e + 16384], 0, 0);
    atomicAdd(&deg[dst[e]], 1.0f);
}

// in-place: deg -> dinv = rsqrt(deg + 1)
__global__ void k_dinv(float* __restrict__ deg, int n_nodes) {
    int i = blockIdx.x * blockDim.x + threadIdx.x;
    if (i < n_nodes) deg[i] = rsqrtf(deg[i] + 1.0f);
}

// ---------------- layer-1 projection: h = x @ W1 via V_WMMA_F32_16X16X4_F32 ----
// One wave computes a 16-row x 16-col (10 valid, rest zero) tile into a
// 16-float padded row. K=256 in 64 steps of 4.
__global__ __launch_bounds__(256) void k_proj1_wmma(
    const float* __restrict__ x, const float* __restrict__ W,
    float* __restrict__ h, int n_nodes) {
    __shared__ float Wpad[IN_F * 16];
    int tid = threadIdx.x;
    for (int i = tid; i < IN_F * 16; i += 256) {
        int k = i >> 4, n = i & 15;
        Wpad[i] = (n < HF) ? W[k * HF + n] : 0.0f;
    }
    __syncthreads();

    int wave = tid >> 5;
    int lane = tid & 31;
    int tile = blockIdx.x * 8 + wave;            // uniform per wave
    int ntiles = (n_nodes + 15) >> 4;
    if (tile >= ntiles) return;                  // whole-wave uniform: EXEC all-1s

    long row0 = (long)tile * 16;
    int mrow  = lane & 15;                       // A row / B col index
    int khalf = (lane >> 4) << 1;                // 0 or 2
    const float* xrow = x + (row0 + mrow) * IN_F + khalf;

    v8f acc = {0.f, 0.f, 0.f, 0.f, 0.f, 0.f, 0.f, 0.f};
#pragma unroll 4
    for (int k = 0; k < IN_F; k += 4) {
        v2f a = *(const v2f*)(xrow + k);         // global_load_b64, 8B aligned
        v2f b;
        b.x = Wpad[(k + khalf) * 16 + mrow];     // ds_load_2addr
        b.y = Wpad[(k + khalf + 1) * 16 + mrow];
        acc = __builtin_amdgcn_wmma_f32_16x16x4_f32(
            /*neg_a=*/false, a, /*neg_b=*/false, b,
            /*c_mod=*/(short)0, acc, /*reuse_a=*/false, /*reuse_b=*/false);
    }

    // C/D layout: VGPR g -> lanes 0-15: M=g, N=lane; lanes 16-31: M=8+g, N=lane-16
    // Padded (16-col) output rows: every lane stores unconditionally, coalesced.
    int outM = (lane >> 4) << 3;                 // 0 or 8
    int outN = lane & 15;
    float* hrow = h + (row0 + outM) * HP + outN;
    if (row0 + 16 <= n_nodes) {
#pragma unroll
        for (int g = 0; g < 8; ++g) hrow[g * HP] = acc[g];
    } else {
#pragma unroll
        for (int g = 0; g < 8; ++g)
            if (row0 + outM + g < n_nodes) hrow[g * HP] = acc[g];
    }
}

// ---------------- edge aggregation: acc[dst] += h[src] * dinv[src]*dinv[dst] ----
// Padded 64B rows: gather = 2x b128 + 1x b64 aligned loads; 10 f32 atomics.
__global__ void k_aggregate(const float* __restrict__ hin, const int* __restrict__ src,
                            const int* __restrict__ dst, const float* __restrict__ dinv,
                            float* __restrict__ acc, int n_edges) {
    int e = blockIdx.x * blockDim.x + threadIdx.x;
    if (e >= n_edges) return;
    __builtin_prefetch(&src[e + 16384], 0, 0);   // global_prefetch_b8 on edge stream
    __builtin_prefetch(&dst[e + 16384], 0, 0);
    int s = src[e], d = dst[e];
    float norm = dinv[s] * dinv[d];              // dinv: 400 KB, L2-resident
    const float* hs = hin + ((long)s << 4);      // 64B-aligned padded row
    v4f a0 = *(const v4f*)(hs);                  // global_load_b128
    v4f a1 = *(const v4f*)(hs + 4);              // global_load_b128
    v2f a2 = *(const v2f*)(hs + 8);              // global_load_b64
    float* ad = acc + ((long)d << 4);
    atomicAdd(&ad[0], a0.x * norm);
    atomicAdd(&ad[1], a0.y * norm);
    atomicAdd(&ad[2], a0.z * norm);
    atomicAdd(&ad[3], a0.w * norm);
    atomicAdd(&ad[4], a1.x * norm);
    atomicAdd(&ad[5], a1.y * norm);
    atomicAdd(&ad[6], a1.z * norm);
    atomicAdd(&ad[7], a1.w * norm);
    atomicAdd(&ad[8], a2.x * norm);
    atomicAdd(&ad[9], a2.y * norm);
}

// ---------------- self-loop + bias + ReLU (in-place on acc, padded rows) ------
__global__ void k_finish(const float* __restrict__ hin, const float* __restrict__ dinv,
                         const float* __restrict__ bias, float* __restrict__ acc, int total) {
    int idx = blockIdx.x * blockDim.x + threadIdx.x;  // total = n_nodes*HP
    if (idx >= total) return;
    int f = idx & (HP - 1);
    if (f >= HF) return;                              // pad cols stay zero
    int i = idx >> 4;
    float di = dinv[i];
    float v = acc[idx] + hin[idx] * di * di + bias[f];
    acc[idx] = v > 0.0f ? v : 0.0f;
}

// ---------------- tiny dense layer 2: hout = hin @ W2 (10x10, padded rows) ----
__global__ void k_proj2(const float* __restrict__ hin, const float* __restrict__ W,
                        float* __restrict__ hout, int total) {
    int idx = blockIdx.x * blockDim.x + threadIdx.x;  // total = n_nodes*HP
    if (idx >= total) return;
    int n = idx & (HP - 1);
    if (n >= HF) return;                              // pad cols never read downstream
    int i = idx >> 4;
    const float* hr = hin + ((long)i << 4);
    float s = 0.0f;
#pragma unroll
    for (int k = 0; k < HF; ++k) s += hr[k] * W[k * HF + n];
    hout[idx] = s;
}

// ---------------- readout: out = h2 @ Wl + bl ----------------
__global__ void k_readout(const float* __restrict__ h2, const float* __restrict__ Wl,
                          const float* __restrict__ bl, float* __restrict__ out, int n_nodes) {
    int i = blockIdx.x * blockDim.x + threadIdx.x;
    if (i >= n_nodes) return;
    const float* hr = h2 + ((long)i << 4);
    float s = bl[0];
#pragma unroll
    for (int f = 0; f < HF; ++f) s += hr[f] * Wl[f];
    out[i] = s;
}

extern "C" void kernel_launch(void* const* d_in, const int* in_sizes, int n_in,
                              void* d_out, int out_size, void* d_ws, size_t ws_size,
                              hipStream_t stream) {
    const float* x  = (const float*)d_in[0];
    const int*   ei = (const int*)  d_in[1];
    const float* W1 = (const float*)d_in[2];
    const float* b1 = (const float*)d_in[3];
    const float* W2 = (const float*)d_in[4];
    const float* b2 = (const float*)d_in[5];
    const float* Wl = (const float*)d_in[6];
    const float* bl = (const float*)d_in[7];
    float* out = (float*)d_out;

    const int n_nodes = in_sizes[0] / IN_F;
    const int n_edges = in_sizes[1] / 2;
    const int* src = ei;
    const int* dst = ei + n_edges;

    // workspace layout: [dinv | acc1 | acc2 | h | h2p], padded rows of HP floats.
    // First three regions are zeroed every launch.
    float* ws   = (float*)d_ws;
    float* dinv = ws;
    float* acc1 = dinv + n_nodes;
    float* acc2 = acc1 + (size_t)n_nodes * HP;
    float* h    = acc2 + (size_t)n_nodes * HP;
    float* h2p  = h    + (size_t)n_nodes * HP;

    const int B = 256;
    const int total = n_nodes * HP;
    const int zn = n_nodes + 2 * total;

    k_zero<<<(zn + B - 1) / B, B, 0, stream>>>(ws, zn);
    k_deg<<<(n_edges + B - 1) / B, B, 0, stream>>>(dst, dinv, n_edges);
    k_dinv<<<(n_nodes + B - 1) / B, B, 0, stream>>>(dinv, n_nodes);

    // layer 1
    int ntiles = (n_nodes + 15) / 16;
    k_proj1_wmma<<<(ntiles + 7) / 8, B, 0, stream>>>(x, W1, h, n_nodes);
    k_aggregate<<<(n_edges + B - 1) / B, B, 0, stream>>>(h, src, dst, dinv, acc1, n_edges);
    k_finish<<<(total + B - 1) / B, B, 0, stream>>>(h, dinv, b1, acc1, total);

    // layer 2
    k_proj2<<<(total + B - 1) / B, B, 0, stream>>>(acc1, W2, h2p, total);
    k_aggregate<<<(n_edges + B - 1) / B, B, 0, stream>>>(h2p, src, dst, dinv, acc2, n_edges);
    k_finish<<<(total + B - 1) / B, B, 0, stream>>>(h2p, dinv, b2, acc2, total);

    // readout
    k_readout<<<(n_nodes + B - 1) / B, B, 0, stream>>>(acc2, Wl, bl, out, n_nodes);
}